// ETGNN_20469814133531
// MI455X (gfx1250) — compile-verified
//
#include <hip/hip_runtime.h>
#include <math.h>

// Problem constants (from reference)
#define BB   32
#define NN   64
#define II   8000
#define DD   64
#define TFDk 4
#define NHOP 2
#define IMPF 0.5f
#define NEGC -1e9f

typedef __attribute__((ext_vector_type(2))) float v2f;
typedef __attribute__((ext_vector_type(4))) float v4f;
typedef __attribute__((ext_vector_type(8))) float v8f;

__global__ __launch_bounds__(128)
void etgnn_fused(const int*   __restrict__ hop_idx,     // (3,32,64)
                 const float* __restrict__ hop_tf,      // (3,32,64,4)
                 const int*   __restrict__ hop_len,     // (3,32)
                 const float* __restrict__ central_tf,  // (32,4)
                 const float* __restrict__ items,       // (8000,64)
                 const float* __restrict__ users,       // (8000,64)
                 const float* __restrict__ semW,        // (3,32)
                 const float* __restrict__ semB,        // (32)
                 const float* __restrict__ perW,        // (1,16)
                 const float* __restrict__ perB,        // (1,16)
                 const float* __restrict__ fcW,         // (128,64)
                 const float* __restrict__ fcB,         // (64)
                 float*       __restrict__ out)         // (32,8000)
{
    __shared__ float q_lds[64 * 64];   // query tile (items rows i0..i0+63)
    __shared__ float e_lds[64 * 64];   // gathered neighbor embeddings (per hop)
    __shared__ float wk_lds[64 * 64];  // fcW_h, then reused as W2 tile
    __shared__ float ta_lds[64];
    __shared__ float bias_lds[64];

    const int tid  = threadIdx.x;
    const int lane = tid & 31;
    const int wave = tid >> 5;        // 4 waves, 16 i-rows each
    const int lx   = lane & 15;       // N column within tile / A-row
    const int hi   = lane >> 4;       // lane half -> K offset 2*hi, M offset 8*hi
    const int b    = blockIdx.y;
    const int i0   = blockIdx.x * 64; // 8000 = 125 * 64, no bounds checks

    // ---- stage query tile (coalesced float4) ----
#pragma unroll
    for (int it = 0; it < 8; ++it) {
        int idx4 = tid + it * 128;
        int row = idx4 >> 4, c4 = (idx4 & 15) << 2;
        *(v4f*)&q_lds[row * 64 + c4] =
            *(const v4f*)&items[(size_t)(i0 + row) * 64 + c4];
    }
    __syncthreads();

    // bias[i] = fc_b . q_i
    if (tid < 64) {
        float s = 0.f;
#pragma unroll
        for (int d = 0; d < 64; ++d) s += fcB[d] * q_lds[tid * 64 + d];
        bias_lds[tid] = s;
    }

    float outc[8];
#pragma unroll
    for (int r = 0; r < 8; ++r) outc[r] = 0.f;

    const int arow = (wave * 16 + lx) * 64 + 2 * hi;  // A-fragment base (Q / W2)

    for (int h = 0; h < NHOP; ++h) {
        const float* table = (h == 0) ? items : users;  // ref hop 1 -> items, hop 2 -> users

        // ---- gather neighbor embedding tile ----
#pragma unroll
        for (int it = 0; it < 8; ++it) {
            int idx4 = tid + it * 128;
            int row = idx4 >> 4, c4 = (idx4 & 15) << 2;
            int node = hop_idx[(h + 1) * BB * NN + b * NN + row];
            *(v4f*)&e_lds[row * 64 + c4] =
                *(const v4f*)&table[(size_t)node * 64 + c4];
        }
        // ---- stage fc_W_h (rows h*64 .. h*64+63) ----
#pragma unroll
        for (int it = 0; it < 8; ++it) {
            int idx4 = tid + it * 128;
            int row = idx4 >> 4, c4 = (idx4 & 15) << 2;
            *(v4f*)&wk_lds[row * 64 + c4] =
                *(const v4f*)&fcW[(size_t)(h * 64 + row) * 64 + c4];
        }
        // ---- temporal attention term ta[b,n] = central_time . time_embed(tf) ----
        if (tid < 64) {
            const float* tf  = hop_tf + (((size_t)(h + 1) * BB + b) * NN + tid) * TFDk;
            const float* ctf = central_tf + b * TFDk;
            float acc = 0.f;
#pragma unroll
            for (int k = 0; k < 32; ++k) {           // semantic halves
                float sn = semB[k], sc = semB[k];
#pragma unroll
                for (int j = 0; j < 3; ++j) {
                    sn += tf[j]  * semW[j * 32 + k];
                    sc += ctf[j] * semW[j * 32 + k];
                }
                acc += sn * sc;
            }
#pragma unroll
            for (int k = 0; k < 16; ++k) {           // periodic halves, scale^2 = 1/16
                float xn = tf[3]  * perW[k] + perB[k];
                float xc = ctf[3] * perW[k] + perB[k];
                acc += 0.0625f * (cosf(xn) * cosf(xc) + sinf(xn) * sinf(xc));
            }
            ta_lds[tid] = acc;
        }
        const int len = hop_len[(h + 1) * BB + b];
        __syncthreads();

        // ---- GEMM 1: W2 tile = Q (64x64) @ fcW_h^T (64x64), via WMMA f32 16x16x4 ----
        v8f accw[4];
#pragma unroll
        for (int t = 0; t < 4; ++t) accw[t] = (v8f){0.f,0.f,0.f,0.f,0.f,0.f,0.f,0.f};
#pragma unroll
        for (int kk = 0; kk < 16; ++kk) {
            int kb = kk * 4;
            v2f A = *(const v2f*)&q_lds[arow + kb];
#pragma unroll
            for (int t = 0; t < 4; ++t) {
                v2f Bv = *(const v2f*)&wk_lds[(t * 16 + lx) * 64 + 2 * hi + kb];
                accw[t] = __builtin_amdgcn_wmma_f32_16x16x4_f32(
                    false, A, false, Bv, (short)0, accw[t], false, false);
            }
        }
        __syncthreads();
        // write W2 tile back to LDS so it can be re-fragmented as an A operand
#pragma unroll
        for (int t = 0; t < 4; ++t)
#pragma unroll
            for (int r = 0; r < 8; ++r)
                wk_lds[(wave * 16 + hi * 8 + r) * 64 + t * 16 + lx] = accw[t][r];
        __syncthreads();

        // ---- GEMM 2: att = Q @ E^T ; val = W2 @ E^T ----
        v8f acca[4], accv[4];
#pragma unroll
        for (int t = 0; t < 4; ++t) {
            acca[t] = (v8f){0.f,0.f,0.f,0.f,0.f,0.f,0.f,0.f};
            accv[t] = (v8f){0.f,0.f,0.f,0.f,0.f,0.f,0.f,0.f};
        }
#pragma unroll
        for (int kk = 0; kk < 16; ++kk) {
            int kb = kk * 4;
            v2f Aq = *(const v2f*)&q_lds[arow + kb];
            v2f Aw = *(const v2f*)&wk_lds[arow + kb];
#pragma unroll
            for (int t = 0; t < 4; ++t) {
                v2f Ev = *(const v2f*)&e_lds[(t * 16 + lx) * 64 + 2 * hi + kb];
                acca[t] = __builtin_amdgcn_wmma_f32_16x16x4_f32(
                    false, Aq, false, Ev, (short)0, acca[t], false, false);
                accv[t] = __builtin_amdgcn_wmma_f32_16x16x4_f32(
                    false, Aw, false, Ev, (short)0, accv[t], false, false);
            }
        }

        // ---- epilogue: +ta, mask, LeakyReLU, softmax over n, weighted reduce ----
        float av[4][8], m[8];
#pragma unroll
        for (int r = 0; r < 8; ++r) m[r] = -INFINITY;
#pragma unroll
        for (int t = 0; t < 4; ++t) {
            int n = t * 16 + lx;
            float add = IMPF * ta_lds[n] + ((n >= len) ? NEGC : 0.f);
#pragma unroll
            for (int r = 0; r < 8; ++r) {
                float a = acca[t][r] + add;
                a = (a >= 0.f) ? a : 0.2f * a;     // LeakyReLU(0.2) after masking
                av[t][r] = a;
                m[r] = fmaxf(m[r], a);
            }
        }
#pragma unroll
        for (int msk = 1; msk <= 8; msk <<= 1)
#pragma unroll
            for (int r = 0; r < 8; ++r)
                m[r] = fmaxf(m[r], __shfl_xor(m[r], msk, 32));  // stays within 16-lane half
        float sum[8], cw[8];
#pragma unroll
        for (int r = 0; r < 8; ++r) { sum[r] = 0.f; cw[r] = 0.f; }
#pragma unroll
        for (int t = 0; t < 4; ++t)
#pragma unroll
            for (int r = 0; r < 8; ++r) {
                float p = __expf(av[t][r] - m[r]);
                sum[r] += p;
                cw[r]  += p * accv[t][r];
            }
#pragma unroll
        for (int msk = 1; msk <= 8; msk <<= 1)
#pragma unroll
            for (int r = 0; r < 8; ++r) {
                sum[r] += __shfl_xor(sum[r], msk, 32);
                cw[r]  += __shfl_xor(cw[r],  msk, 32);
            }
#pragma unroll
        for (int r = 0; r < 8; ++r) outc[r] += cw[r] / sum[r];
        __syncthreads();   // protect e_lds / wk_lds / ta_lds before next hop
    }

    if (lx == 0) {
#pragma unroll
        for (int r = 0; r < 8; ++r) {
            int row = wave * 16 + hi * 8 + r;
            out[(size_t)b * II + i0 + row] = outc[r] + bias_lds[row];
        }
    }
}

extern "C" void kernel_launch(void* const* d_in, const int* in_sizes, int n_in,
                              void* d_out, int out_size, void* d_ws, size_t ws_size,
                              hipStream_t stream) {
    (void)in_sizes; (void)n_in; (void)out_size; (void)d_ws; (void)ws_size;
    etgnn_fused<<<dim3(II / 64, BB, 1), 128, 0, stream>>>(
        (const int*)d_in[0],   (const float*)d_in[1], (const int*)d_in[2],
        (const float*)d_in[3], (const float*)d_in[4], (const float*)d_in[5],
        (const float*)d_in[6], (const float*)d_in[7], (const float*)d_in[8],
        (const float*)d_in[9], (const float*)d_in[10], (const float*)d_in[11],
        (float*)d_out);
}